// GNN_35699768165187
// MI455X (gfx1250) — compile-verified
//
#include <hip/hip_runtime.h>
#include <hip/hip_bf16.h>

typedef float v2f __attribute__((ext_vector_type(2)));
typedef float v8f __attribute__((ext_vector_type(8)));

// ---------------- workspace zeroing (deterministic across graph replays) ----
__global__ void GNN_zero_kernel(float* __restrict__ p, long n) {
    long i = (long)blockIdx.x * blockDim.x + threadIdx.x;
    long stride = (long)gridDim.x * blockDim.x;
    for (; i < n; i += stride) p[i] = 0.0f;
}

// ---------------- deg[col[e]] += w[e] ---------------------------------------
__global__ void GNN_deg_kernel(const long long* __restrict__ col,
                               const float* __restrict__ w,
                               float* __restrict__ deg, long E) {
    long i = (long)blockIdx.x * blockDim.x + threadIdx.x;
    if (i >= E) return;
    int c = (int)col[i];
    unsafeAtomicAdd(&deg[c], w[i]);   // global_atomic_add_f32, L2-resident (4 MB)
}

// ---------------- deg -> d^{-1/2} in place ----------------------------------
__global__ void GNN_dinv_kernel(float* __restrict__ deg, long N) {
    long i = (long)blockIdx.x * blockDim.x + threadIdx.x;
    if (i >= N) return;
    float d = deg[i];
    deg[i] = (d > 0.0f) ? __frsqrt_rn(d) : 0.0f;
}

// ---------------- one propagation hop: hout[col] += hin[row] * norm ---------
// norm recomputed on the fly (dinv gathers hit L2) to avoid 256MB of HBM
// traffic for a materialized norm array.
__global__ void GNN_hop_kernel(const long long* __restrict__ row,
                               const long long* __restrict__ col,
                               const float* __restrict__ w,
                               const float* __restrict__ dinv,
                               const float* __restrict__ hin,
                               float* __restrict__ hout, long E) {
    long i = (long)blockIdx.x * blockDim.x + threadIdx.x;
    if (i >= E) return;
    int r = (int)row[i];
    int c = (int)col[i];
    float nm = dinv[r] * w[i] * dinv[c];
    const float2 hv = *(const float2*)(hin + 2ll * r);   // 8B L2-resident gather
    unsafeAtomicAdd(&hout[2ll * c + 0], hv.x * nm);
    unsafeAtomicAdd(&hout[2ll * c + 1], hv.y * nm);
}

// ---------------- final linear + ReLU via V_WMMA_F32_16X16X4_F32 ------------
// Each wave computes 16 nodes: D = A(16x4) x B(4x16) + C, twice (feats 4+2).
// A layout per ISA 32-bit 16x4 table: lanes 0-15 give K={0,1} in the two
// A VGPRs, lanes 16-31 give K={2,3}.  B = W broadcast across all 16 columns,
// so every column of the accumulator equals the per-node output.
__global__ void GNN_final_kernel(const float* __restrict__ x,
                                 const float* __restrict__ h1,
                                 const float* __restrict__ h2,
                                 const float* __restrict__ Wp,
                                 const float* __restrict__ bp,
                                 float* __restrict__ out, int N) {
    const int lane = threadIdx.x & 31;
    const int wid  = blockIdx.x * (blockDim.x >> 5) + (threadIdx.x >> 5);
    const int base = wid * 16;
    if (base >= N) return;

    const float W0 = Wp[0], W1 = Wp[1], W2 = Wp[2];
    const float W3 = Wp[3], W4 = Wp[4], W5 = Wp[5];
    const float bias = bp[0];

    if (base + 16 <= N) {                    // full tile: wave-uniform, EXEC all 1s
        const bool hi = lane >= 16;          // lanes 16-31 supply K=2,3
        const int  m  = base + (lane & 15);  // node row M = lane % 16

        // WMMA #1: feats K=0..3 = {x0, x1, h1_0, h1_1}
        const float* s1 = hi ? h1 : x;
        v2f a1; a1[0] = s1[2 * m]; a1[1] = s1[2 * m + 1];
        v2f b1; b1[0] = hi ? W2 : W0; b1[1] = hi ? W3 : W1;

        // WMMA #2: feats K=0..3 = {h2_0, h2_1, 0, 0}
        v2f a2; a2[0] = hi ? 0.0f : h2[2 * m];
                a2[1] = hi ? 0.0f : h2[2 * m + 1];
        v2f b2; b2[0] = hi ? 0.0f : W4; b2[1] = hi ? 0.0f : W5;

        v8f c = {};
        c = __builtin_amdgcn_wmma_f32_16x16x4_f32(false, a1, false, b1,
                                                  (short)0, c, false, false);
        c = __builtin_amdgcn_wmma_f32_16x16x4_f32(false, a2, false, b2,
                                                  (short)0, c, false, false);

        // D layout: VGPR r holds row M=r (lanes 0-15) and M=r+8 (lanes 16-31);
        // B was constant across N, so every lane's c[r] == out[M].
        int r = -1, o = 0;
        if (lane < 8)                     { r = lane;      o = base + lane;     }
        else if (lane >= 16 && lane < 24) { r = lane - 16; o = base + 8 + r;    }
        if (r >= 0) {
            float v = c[0];
            #pragma unroll
            for (int k = 1; k < 8; ++k) if (r == k) v = c[k];
            out[o] = fmaxf(v + bias, 0.0f);
        }
    } else {                                 // partial tail tile: scalar path
        const int m = base + lane;
        if (lane < 16 && m < N) {
            float v = W0 * x[2 * m]  + W1 * x[2 * m + 1]
                    + W2 * h1[2 * m] + W3 * h1[2 * m + 1]
                    + W4 * h2[2 * m] + W5 * h2[2 * m + 1] + bias;
            out[m] = fmaxf(v, 0.0f);
        }
    }
}

extern "C" void kernel_launch(void* const* d_in, const int* in_sizes, int n_in,
                              void* d_out, int out_size, void* d_ws, size_t ws_size,
                              hipStream_t stream) {
    const float*     x  = (const float*)d_in[0];
    const long long* ei = (const long long*)d_in[1];   // int64 per reference
    const float*     w  = (const float*)d_in[2];
    const float*     W  = (const float*)d_in[3];
    const float*     b  = (const float*)d_in[4];
    float*           out = (float*)d_out;

    const long N = (long)in_sizes[0] / 2;   // x is [N,2]
    const long E = (long)in_sizes[1] / 2;   // edge_index is [2,E]

    const long long* row = ei;
    const long long* col = ei + E;

    // workspace: deg/dinv [N] | h1 [2N] | h2 [2N]  (20 MB @ N=1e6)
    float* ws  = (float*)d_ws;
    float* deg = ws;
    float* h1  = ws + N;
    float* h2  = ws + 3 * N;

    const int B = 256;
    const long edgeBlocks = (E + B - 1) / B;
    const long nodeBlocks = (N + B - 1) / B;

    GNN_zero_kernel<<<4096, B, 0, stream>>>(ws, 5 * N);
    GNN_deg_kernel<<<(unsigned)edgeBlocks, B, 0, stream>>>(col, w, deg, E);
    GNN_dinv_kernel<<<(unsigned)nodeBlocks, B, 0, stream>>>(deg, N);
    GNN_hop_kernel<<<(unsigned)edgeBlocks, B, 0, stream>>>(row, col, w, deg, x,  h1, E);
    GNN_hop_kernel<<<(unsigned)edgeBlocks, B, 0, stream>>>(row, col, w, deg, h1, h2, E);

    const long waves  = (N + 15) / 16;            // 16 nodes per wave32
    const long blocks = (waves + 7) / 8;          // 8 waves per 256-thread block
    GNN_final_kernel<<<(unsigned)blocks, B, 0, stream>>>(x, h1, h2, W, b, out, (int)N);
}